// CGNN_22419729285625
// MI455X (gfx1250) — compile-verified
//
#include <hip/hip_runtime.h>

// MI455X / gfx1250 fused CGNN kernel.
// Roofline: ~3.3 GFLOP, ~90 MB mandatory HBM traffic (~3.9us at 23.3 TB/s).
// Fully fused: h/pe tiles + 32-ch intermediate live in LDS; the 6->32
// pointwise GEMM runs on the fp32 matrix pipe (V_WMMA_F32_16X16X4_F32);
// depthwise 3x3s run on VALU with float4-vectorized LDS traffic.

typedef float v2f __attribute__((ext_vector_type(2)));
typedef float v8f __attribute__((ext_vector_type(8)));

#define B_    16
#define S_    768
#define T_    14
#define A_    16
#define HK0_  6
#define PEK0_ 2
#define U_    32
#define K1_   2

#define TS   16          // output s-rows per workgroup (S_ % TS == 0)
#define ST1  (TS + 2)    // 18: rows where post-relu x is materialized (halo 1)
#define TT1  (T_ + 2)    // 16: cols of x tile (halo 1)
#define ST2  (TS + 4)    // 20: rows of raw h / pe tile (halo 2)
#define TT2  (T_ + 4)    // 18: cols of raw h / pe tile

#if __has_builtin(__builtin_amdgcn_wmma_f32_16x16x4_f32)
#define HAVE_WMMA_F32X4 1
#else
#define HAVE_WMMA_F32X4 0
#endif

__global__ __launch_bounds__(256)
void cgnn_fused(const float* __restrict__ h1, const float* __restrict__ h2,
                const float* __restrict__ pe,
                const float* __restrict__ w_hh, const float* __restrict__ b_hh,
                const float* __restrict__ w_vh, const float* __restrict__ b_vh,
                const float* __restrict__ w_hp, const float* __restrict__ b_hp,
                const float* __restrict__ w_vp, const float* __restrict__ b_vp,
                const float* __restrict__ w_oh, const float* __restrict__ b_oh,
                const float* __restrict__ w_ov, const float* __restrict__ b_ov,
                float* __restrict__ out)
{
    // LDS: 8640 + 2880 + 36864 + ~3.1KB weights ~= 51.5 KB (< 64KB static)
    __shared__ float hT [ST2][TT2][HK0_];
    __shared__ float peT[ST2][TT2][PEK0_];
    __shared__ float xT [ST1][TT1][U_];      // holds pe-branch p, then relu'd x
    __shared__ float sw_hh[9*HK0_],  sb_hh[HK0_];
    __shared__ float sw_vh[HK0_*U_], sb_vh[U_];
    __shared__ float sw_hp[9*PEK0_], sb_hp[PEK0_];
    __shared__ float sw_vp[PEK0_*U_],sb_vp[U_];
    __shared__ float sw_oh[9*U_],    sb_oh[U_];
    __shared__ float sw_ov[U_*K1_],  sb_ov[K1_];

    const int tid = threadIdx.x;
    const int nba = blockIdx.x;          // b*A + a
    const int b   = nba >> 4;
    const int a   = nba & 15;
    const int s0  = blockIdx.y * TS;

    // ---- stage 0: stage all weights/biases into LDS -------------------------
    for (int i = tid; i < 9*HK0_;   i += 256) sw_hh[i] = w_hh[i];
    for (int i = tid; i < HK0_;     i += 256) sb_hh[i] = b_hh[i];
    for (int i = tid; i < HK0_*U_;  i += 256) sw_vh[i] = w_vh[i];
    for (int i = tid; i < U_;       i += 256) sb_vh[i] = b_vh[i];
    for (int i = tid; i < 9*PEK0_;  i += 256) sw_hp[i] = w_hp[i];
    for (int i = tid; i < PEK0_;    i += 256) sb_hp[i] = b_hp[i];
    for (int i = tid; i < PEK0_*U_; i += 256) sw_vp[i] = w_vp[i];
    for (int i = tid; i < U_;       i += 256) sb_vp[i] = b_vp[i];
    for (int i = tid; i < 9*U_;     i += 256) sw_oh[i] = w_oh[i];
    for (int i = tid; i < U_;       i += 256) sb_oh[i] = b_oh[i];
    for (int i = tid; i < U_*K1_;   i += 256) sw_ov[i] = w_ov[i];
    for (int i = tid; i < K1_;      i += 256) sb_ov[i] = b_ov[i];

    // ---- stage 1: load h (= concat(h2,h1)) and pe tiles, zero halo ----------
    // One position per thread so the 3-float reads coalesce into b96 loads.
    for (int q = tid; q < ST2*TT2; q += 256) {
        const int tt = q % TT2;
        const int ss = q / TT2;
        const int s  = s0 - 2 + ss;
        const int t  = tt - 2;
        float v0 = 0.f, v1 = 0.f, v2 = 0.f, v3 = 0.f, v4 = 0.f, v5 = 0.f;
        float p0 = 0.f, p1 = 0.f;
        if (s >= 0 && s < S_ && t >= 0 && t < T_) {
            const size_t base = ((((size_t)b * S_ + s) * T_ + t) * A_ + a) * 3;
            const float* q2 = h2 + base;
            const float* q1 = h1 + base;
            v0 = q2[0]; v1 = q2[1]; v2 = q2[2];
            v3 = q1[0]; v4 = q1[1]; v5 = q1[2];
            const float* qp = pe + (((size_t)b * S_ + s) * T_ + t) * PEK0_;
            p0 = qp[0]; p1 = qp[1];
        }
        float* hp = &hT[ss][tt][0];
        hp[0] = v0; hp[1] = v1; hp[2] = v2; hp[3] = v3; hp[4] = v4; hp[5] = v5;
        peT[ss][tt][0] = p0; peT[ss][tt][1] = p1;
    }

    // prefetch next s-tile into L2 (global_prefetch_b8)
    if (blockIdx.y + 1u < gridDim.y) {
        const int sN = s0 + TS;
        for (int q = tid; q < TS*T_; q += 256) {
            const int tt = q % T_, st = q / T_;
            const size_t gi = ((((size_t)b * S_ + (sN + st)) * T_ + tt) * A_ + a) * 3;
            __builtin_prefetch(h1 + gi, 0, 1);
            __builtin_prefetch(h2 + gi, 0, 1);
        }
    }
    __syncthreads();

    // ---- stage 2: pe branch (dw3x3 2ch + 1x1 2->32) into xT -----------------
    for (int q = tid; q < ST1*TT1; q += 256) {
        const int tt2 = q & 15;          // TT1 == 16
        const int ss2 = q >> 4;
        float p0 = sb_hp[0], p1 = sb_hp[1];
#pragma unroll
        for (int di = 0; di < 3; ++di)
#pragma unroll
            for (int dj = 0; dj < 3; ++dj) {
                const float* pp = &peT[ss2 + di][tt2 + dj][0];
                p0 = fmaf(pp[0], sw_hp[(di*3+dj)*2 + 0], p0);
                p1 = fmaf(pp[1], sw_hp[(di*3+dj)*2 + 1], p1);
            }
        float4* xp = (float4*)&xT[ss2][tt2][0];   // 128B-aligned slice
#pragma unroll
        for (int c4 = 0; c4 < 8; ++c4) {
            const float4 w0 = *(const float4*)&sw_vp[c4*4];
            const float4 w1 = *(const float4*)&sw_vp[U_ + c4*4];
            const float4 bv = *(const float4*)&sb_vp[c4*4];
            float4 r;
            r.x = fmaf(p0, w0.x, fmaf(p1, w1.x, bv.x));
            r.y = fmaf(p0, w0.y, fmaf(p1, w1.y, bv.y));
            r.z = fmaf(p0, w0.z, fmaf(p1, w1.z, bv.z));
            r.w = fmaf(p0, w0.w, fmaf(p1, w1.w, bv.w));
            xp[c4] = r;
        }
    }
    __syncthreads();

    // ---- stage 3: dw3x3(6ch) + WMMA 1x1 6->32, add pe, relu, back to xT -----
    const int lane  = tid & 31;
    const int wv    = tid >> 5;          // 8 waves
    const int mlane = lane & 15;         // M position / N column within tile
    const int khalf = lane >> 4;         // K lane-half per ISA A/B layout

#if HAVE_WMMA_F32X4
    // B fragments (K x N): lane holds B[k = 2*khalf + j][n = mlane (+16)], j=0,1
    const int kc0 = khalf * 2;           // chunk0 -> channels 0..3
    v2f B00 = { sw_vh[kc0*U_ + mlane],       sw_vh[(kc0+1)*U_ + mlane] };
    v2f B10 = { sw_vh[kc0*U_ + mlane + 16],  sw_vh[(kc0+1)*U_ + mlane + 16] };
    // chunk1 -> channels 4,5 from lane-half 0; zero pad (ch 6,7) from half 1
    v2f B01 = { khalf ? 0.f : sw_vh[4*U_ + mlane],
                khalf ? 0.f : sw_vh[5*U_ + mlane] };
    v2f B11 = { khalf ? 0.f : sw_vh[4*U_ + mlane + 16],
                khalf ? 0.f : sw_vh[5*U_ + mlane + 16] };
#endif

    for (int g = wv; g < ST1; g += 8) {          // wave-uniform: EXEC all 1s
        const int ss2 = g;
        const int tt2 = mlane;                   // this lane's position (M index)
        float c6[HK0_];
#pragma unroll
        for (int c = 0; c < HK0_; ++c) c6[c] = sb_hh[c];
#pragma unroll
        for (int di = 0; di < 3; ++di)
#pragma unroll
            for (int dj = 0; dj < 3; ++dj) {
                const float* hp = &hT[ss2 + di][tt2 + dj][0];
                const float* wp = &sw_hh[(di*3+dj)*HK0_];
#pragma unroll
                for (int c = 0; c < HK0_; ++c) c6[c] = fmaf(hp[c], wp[c], c6[c]);
            }

        const int spos = s0 - 1 + ss2;

#if HAVE_WMMA_F32X4
        // A fragments (M x K): lane holds A[m=mlane][k = 2*khalf + j].
        // Explicit khalf selects (1 cndmask each) -- no runtime array indexing.
        v2f A0 = { khalf ? c6[2] : c6[0], khalf ? c6[3] : c6[1] };
        v2f A1 = { khalf ? 0.f   : c6[4], khalf ? 0.f   : c6[5] };
        v8f acc0 = {0.f,0.f,0.f,0.f,0.f,0.f,0.f,0.f};
        v8f acc1 = {0.f,0.f,0.f,0.f,0.f,0.f,0.f,0.f};
        acc0 = __builtin_amdgcn_wmma_f32_16x16x4_f32(false, A0, false, B00, (short)0, acc0, false, false);
        acc0 = __builtin_amdgcn_wmma_f32_16x16x4_f32(false, A1, false, B01, (short)0, acc0, false, false);
        acc1 = __builtin_amdgcn_wmma_f32_16x16x4_f32(false, A0, false, B10, (short)0, acc1, false, false);
        acc1 = __builtin_amdgcn_wmma_f32_16x16x4_f32(false, A1, false, B11, (short)0, acc1, false, false);

        // Branchless writeback. s-boundary is wave-uniform; t-boundary hits
        // only (j==0,khalf==0) -> m=0 and (j==7,khalf==1) -> m=15.
        const float sMsk  = ((spos >= 0) && (spos < S_)) ? 1.f : 0.f;
        const float loMsk = khalf ? 1.f : 0.f;
        const float hiMsk = khalf ? 0.f : 1.f;
        const float bv0   = sb_vh[mlane];
        const float bv1   = sb_vh[mlane + 16];
#pragma unroll
        for (int j = 0; j < 8; ++j) {
            const int m = j + 8*khalf;           // output column (tt2 of x)
            float msk = sMsk;
            if (j == 0) msk *= loMsk;            // folded at unroll time
            if (j == 7) msk *= hiMsk;
            float* xp = &xT[ss2][m][mlane];
            const float v0 = acc0[j] + bv0 + xp[0];
            const float v1 = acc1[j] + bv1 + xp[16];
            xp[0]  = fmaxf(v0, 0.f) * msk;
            xp[16] = fmaxf(v1, 0.f) * msk;
        }
#else
        // VALU fallback (kept so a missing builtin still compiles)
        const int tpos = tt2 - 1;
        const float msk = ((spos >= 0) && (spos < S_) && (tpos >= 0) && (tpos < T_)) ? 1.f : 0.f;
#pragma unroll
        for (int nn = 0; nn < 16; ++nn) {
            const int nidx = nn + 16 * khalf;
            float acc = sb_vh[nidx];
#pragma unroll
            for (int c = 0; c < HK0_; ++c) acc = fmaf(c6[c], sw_vh[c*U_ + nidx], acc);
            acc += xT[ss2][tt2][nidx];
            xT[ss2][tt2][nidx] = fmaxf(acc, 0.f) * msk;
        }
#endif
    }
    __syncthreads();

    // ---- stage 4: dw3x3 on 32ch + 1x1 32->2, write [B,S,T,A,2] --------------
    // Tap-outer / channel-vectorized: 72 ds_load_b128 instead of 288 b32.
    for (int q = tid; q < TS*T_; q += 256) {
        const int tt = q % T_;
        const int st = q / T_;
        float accv[U_];
#pragma unroll
        for (int c = 0; c < U_; ++c) accv[c] = sb_oh[c];
#pragma unroll
        for (int di = 0; di < 3; ++di)
#pragma unroll
            for (int dj = 0; dj < 3; ++dj) {
                const float* xp = &xT[st + di][tt + dj][0];     // 128B aligned
                const float* wp = &sw_oh[(di*3+dj)*U_];
#pragma unroll
                for (int c4 = 0; c4 < 8; ++c4) {
                    const float4 xv = *(const float4*)(xp + c4*4);
                    const float4 wvv = *(const float4*)(wp + c4*4);
                    accv[c4*4+0] = fmaf(xv.x, wvv.x, accv[c4*4+0]);
                    accv[c4*4+1] = fmaf(xv.y, wvv.y, accv[c4*4+1]);
                    accv[c4*4+2] = fmaf(xv.z, wvv.z, accv[c4*4+2]);
                    accv[c4*4+3] = fmaf(xv.w, wvv.w, accv[c4*4+3]);
                }
            }
        float y0 = sb_ov[0], y1 = sb_ov[1];
#pragma unroll
        for (int c = 0; c < U_; ++c) {
            y0 = fmaf(accv[c], sw_ov[c*2 + 0], y0);
            y1 = fmaf(accv[c], sw_ov[c*2 + 1], y1);
        }
        const size_t o = ((((size_t)b * S_ + (s0 + st)) * T_ + tt) * A_ + a) * K1_;
        out[o + 0] = y0;
        out[o + 1] = y1;
    }
}

extern "C" void kernel_launch(void* const* d_in, const int* in_sizes, int n_in,
                              void* d_out, int out_size, void* d_ws, size_t ws_size,
                              hipStream_t stream) {
    (void)in_sizes; (void)n_in; (void)out_size; (void)d_ws; (void)ws_size;
    const float* h1   = (const float*)d_in[0];
    const float* h2   = (const float*)d_in[1];
    const float* pe   = (const float*)d_in[2];
    const float* w_hh = (const float*)d_in[3];
    const float* b_hh = (const float*)d_in[4];
    const float* w_vh = (const float*)d_in[5];
    const float* b_vh = (const float*)d_in[6];
    const float* w_hp = (const float*)d_in[7];
    const float* b_hp = (const float*)d_in[8];
    const float* w_vp = (const float*)d_in[9];
    const float* b_vp = (const float*)d_in[10];
    const float* w_oh = (const float*)d_in[11];
    const float* b_oh = (const float*)d_in[12];
    const float* w_ov = (const float*)d_in[13];
    const float* b_ov = (const float*)d_in[14];
    float* out = (float*)d_out;

    dim3 grid(B_ * A_, S_ / TS);   // 256 x 48 workgroups
    cgnn_fused<<<grid, 256, 0, stream>>>(h1, h2, pe, w_hh, b_hh, w_vh, b_vh,
                                         w_hp, b_hp, w_vp, b_vp, w_oh, b_oh,
                                         w_ov, b_ov, out);
}